// GCN_3831110828646
// MI455X (gfx1250) — compile-verified
//
#include <hip/hip_runtime.h>
#include <hip/hip_bf16.h>

typedef __attribute__((ext_vector_type(2))) float v2f;
typedef __attribute__((ext_vector_type(8))) float v8f;

// ---------------------------------------------------------------------------
// Degree / normalization
// ---------------------------------------------------------------------------
__global__ void zero_i32_kernel(int* __restrict__ p, int n) {
    int i = blockIdx.x * blockDim.x + threadIdx.x;
    if (i < n) p[i] = 0;
}

__global__ void count_deg_kernel(const int* __restrict__ dst, int* __restrict__ deg, int E) {
    int e = blockIdx.x * blockDim.x + threadIdx.x;
    if (e < E) atomicAdd(&deg[dst[e]], 1);
}

__global__ void calc_dinv_kernel(const int* __restrict__ deg, float* __restrict__ dinv, int n) {
    int i = blockIdx.x * blockDim.x + threadIdx.x;
    if (i < n) {
        // self-loop contributes +1; deg+1 >= 1 always, matching reference where()
        dinv[i] = rsqrtf((float)(deg[i] + 1));
    }
}

// ---------------------------------------------------------------------------
// fp32 WMMA GEMM: C[M x ncols_padded] = (RELU?)(A[M x K]) * B[K x ncols]
// One wave per 16x16 output tile.  V_WMMA_F32_16X16X4_F32, K stepped by 4.
// A frag (16x4, 2 VGPR): lanes 0-15 hold K={0,1}, lanes 16-31 hold K={2,3}.
// B frag (4x16, 2 VGPR): v0 = rows K=0(l0-15)/K=1(l16-31), v1 = K=2/K=3.
// D (16x16, 8 VGPR): VGPR i -> row i (lanes 0-15) / row i+8 (lanes 16-31).
// Column guard is branch-free: clamp to a valid column, scale frag by 0/1.
// ---------------------------------------------------------------------------
template <bool RELU>
__global__ __launch_bounds__(32) void gemm_wmma_f32_kernel(
    const float* __restrict__ A, int lda,
    const float* __restrict__ B, int ldb, int ncols,
    float* __restrict__ C, int ldc,
    int K)
{
    const int lane = threadIdx.x;       // 0..31, wave32
    const int half = lane >> 4;         // 0 or 1
    const int l    = lane & 15;
    const long row0 = (long)blockIdx.x * 16;
    const long col0 = (long)blockIdx.y * 16;
    const int  col  = (int)col0 + l;

    const int   ccol  = (col < ncols) ? col : (ncols - 1);   // always-valid load
    const float bmask = (col < ncols) ? 1.0f : 0.0f;         // zero padded cols

    const float* __restrict__ aptr  = A + (row0 + l) * (long)lda + 2 * half;
    const float* __restrict__ b0ptr = B + (long)half * ldb + ccol;        // rows K+0/K+1
    const float* __restrict__ b1ptr = B + (long)(2 + half) * ldb + ccol;  // rows K+2/K+3
    const long bstep = 4L * ldb;

    v8f acc = {};
#pragma unroll 4
    for (int k = 0; k < K; k += 4) {
        v2f a;
        a.x = aptr[0];
        a.y = aptr[1];
        if (RELU) { a.x = fmaxf(a.x, 0.0f); a.y = fmaxf(a.y, 0.0f); }

        v2f b;
        b.x = b0ptr[0] * bmask;
        b.y = b1ptr[0] * bmask;

        acc = __builtin_amdgcn_wmma_f32_16x16x4_f32(
            /*neg_a=*/false, a, /*neg_b=*/false, b,
            /*c_mod=*/(short)0, acc, /*reuse_a=*/false, /*reuse_b=*/false);

        aptr  += 4;
        b0ptr += bstep;
        b1ptr += bstep;
    }

    float* Cbase = C + (row0 + 8 * half) * (long)ldc + col0 + l;
#pragma unroll
    for (int i = 0; i < 8; ++i) {
        Cbase[(long)i * ldc] = acc[i];   // padded cols (>= ncols) store exact 0
    }
}

// ---------------------------------------------------------------------------
// Aggregation: out = self-loop term + bias, then atomic scatter over edges
// ---------------------------------------------------------------------------
__global__ void init_agg_kernel(const float* __restrict__ xw,
                                const float* __restrict__ dinv,
                                const float* __restrict__ bias,
                                float* __restrict__ out,
                                long total, int LD, int Freal)
{
    long idx = (long)blockIdx.x * blockDim.x + threadIdx.x;
    if (idx >= total) return;
    int  f = (int)(idx % LD);
    long i = idx / LD;
    float di = dinv[i];
    float bs = (f < Freal) ? bias[f] : 0.0f;
    out[idx] = xw[idx] * di * di + bs;
}

// warp-per-edge scatter-add; destination rows are L2-resident (<=52 MB of 192 MB)
__global__ __launch_bounds__(256) void edge_agg_kernel(
    const int* __restrict__ src, const int* __restrict__ dst,
    const float* __restrict__ dinv,
    const float* __restrict__ xw, float* __restrict__ out,
    int E, int F, int LD)
{
    int e = blockIdx.x * blockDim.y + threadIdx.y;
    if (e >= E) return;
    int s = src[e];
    int d = dst[e];
    float nrm = dinv[s] * dinv[d];
    const float* xrow = xw + (long)s * LD;
    float*       orow = out + (long)d * LD;
    for (int f = threadIdx.x; f < F; f += 32) {
        atomicAdd(&orow[f], xrow[f] * nrm);
    }
}

// ---------------------------------------------------------------------------
// Final linear: out[n] = sum_f h2[n,f] * Wl[f] + bl   (wave per node)
// ---------------------------------------------------------------------------
__global__ __launch_bounds__(256) void final_linear_kernel(
    const float* __restrict__ h2, const float* __restrict__ Wl,
    const float* __restrict__ bl, float* __restrict__ out,
    int N, int F, int LD)
{
    int wid  = threadIdx.x >> 5;
    int lane = threadIdx.x & 31;
    long n = (long)blockIdx.x * 8 + wid;
    if (n >= N) return;
    const float* row = h2 + n * (long)LD;
    float sum = 0.0f;
    for (int f = lane; f < F; f += 32) sum += row[f] * Wl[f];
#pragma unroll
    for (int off = 16; off > 0; off >>= 1) sum += __shfl_down(sum, off, 32);
    if (lane == 0) out[n] = sum + bl[0];
}

// ---------------------------------------------------------------------------
// Host-side launch
// ---------------------------------------------------------------------------
extern "C" void kernel_launch(void* const* d_in, const int* in_sizes, int n_in,
                              void* d_out, int out_size, void* d_ws, size_t ws_size,
                              hipStream_t stream)
{
    const float* x  = (const float*)d_in[0];
    const int*   ei = (const int*)d_in[1];
    const float* W1 = (const float*)d_in[2];
    const float* b1 = (const float*)d_in[3];
    const float* W2 = (const float*)d_in[4];
    const float* b2 = (const float*)d_in[5];
    const float* Wl = (const float*)d_in[6];
    const float* bl = (const float*)d_in[7];

    const int N   = 100000;
    const int F0  = 64;
    const int H   = 128;    // hidden, also stride of layer-1 buffers
    const int O   = 100;    // conv2 out
    const int LD2 = 112;    // padded stride for layer-2 buffers (multiple of 16)
    const int E   = in_sizes[1] / 2;
    const int* src = ei;
    const int* dst = ei + E;

    float* ws   = (float*)d_ws;
    float* dinv = ws;                                  // N floats
    int*   deg  = (int*)(ws + N);                      // N ints
    float* xw   = ws + 2L * N;                         // N*H floats (reused L2: N*LD2)
    float* h    = ws + 2L * N + (long)N * H;           // N*H floats (reused L2: N*LD2)
    float* out  = (float*)d_out;

    // 1. degrees + dinv
    zero_i32_kernel<<<(N + 255) / 256, 256, 0, stream>>>(deg, N);
    count_deg_kernel<<<(E + 255) / 256, 256, 0, stream>>>(dst, deg, E);
    calc_dinv_kernel<<<(N + 255) / 256, 256, 0, stream>>>(deg, dinv, N);

    // 2. xw1 = x @ W1   (M=100000=6250*16, K=64, N=128)
    {
        dim3 grid(N / 16, H / 16);
        gemm_wmma_f32_kernel<false><<<grid, 32, 0, stream>>>(x, F0, W1, H, H, xw, H, F0);
    }

    // 3. h1 = scatter-agg(xw1) + b1
    {
        long total = (long)N * H;
        init_agg_kernel<<<(int)((total + 255) / 256), 256, 0, stream>>>(
            xw, dinv, b1, h, total, H, H);
        dim3 blk(32, 8);
        edge_agg_kernel<<<(E + 7) / 8, blk, 0, stream>>>(src, dst, dinv, xw, h, E, H, H);
    }

    // 4. xw2 = relu(h1) @ W2   (K=128, ncols=100 padded to 112)
    {
        dim3 grid(N / 16, LD2 / 16);   // 6250 x 7 tiles
        gemm_wmma_f32_kernel<true><<<grid, 32, 0, stream>>>(h, H, W2, O, O, xw, LD2, H);
    }

    // 5. h2 = scatter-agg(xw2) + b2   (stride LD2, active cols = 100)
    {
        long total = (long)N * LD2;
        init_agg_kernel<<<(int)((total + 255) / 256), 256, 0, stream>>>(
            xw, dinv, b2, h, total, LD2, O);
        dim3 blk(32, 8);
        edge_agg_kernel<<<(E + 7) / 8, blk, 0, stream>>>(src, dst, dinv, xw, h, E, O, LD2);
    }

    // 6. out = h2 @ Wl + bl
    final_linear_kernel<<<(N + 7) / 8, 256, 0, stream>>>(h, Wl, bl, out, N, O, LD2);
}